// SelectiveSSMLayer_41240275976483
// MI455X (gfx1250) — compile-verified
//
#include <hip/hip_runtime.h>
#include <math.h>

// SelectiveSSMLayer forward for MI455X (gfx1250), wave32 + FP32 WMMA.
//   1) xp   = x @ W_in^T                        (wmma f32 16x16x4, sw-pipelined)
//   2) xc   = gelu(xp[:,:512]) + conv(xp[:,512:]) + conv_b   (3 shifted wmma GEMMs)
//   3) cwr/dw/g = needed slices of ssm/gate GEMMs (unused slices skipped)
//   4) xout = xc*dw^2 ; cwgt = g*cwr
//   5) fused scan+contract with async global->LDS double-buffered staging of
//      u[b,t,:], C_weight[b,t,:] shared by the 8 waves of a workgroup
//   6) opre = y @ W_out^T
//   7) row LayerNorm(256) * z + residual x

typedef float v2f __attribute__((ext_vector_type(2)));
typedef float v8f __attribute__((ext_vector_type(8)));
typedef int   v2i __attribute__((vector_size(8)));   // matches async b64 builtin

#define LB 256   // sequence length
#define IB 512   // inner dim I == S
#define HB 256   // model dim

// ---- async global->LDS (ASYNCcnt) path, guarded -----------------------------
#if defined(__has_builtin)
#  if __has_builtin(__builtin_amdgcn_global_load_async_to_lds_b64) && \
      __has_builtin(__builtin_amdgcn_s_wait_asynccnt)
#    define USE_ASYNC_LDS 1
#  endif
#endif
#ifndef USE_ASYNC_LDS
#  define USE_ASYNC_LDS 0
#endif

__device__ __forceinline__ void async_copy_b64(const float* g, float* l) {
#if USE_ASYNC_LDS
  __builtin_amdgcn_global_load_async_to_lds_b64(
      (__attribute__((address_space(1))) v2i*)(g),
      (__attribute__((address_space(3))) v2i*)(l), 0, 0);
#else
  *(v2f*)l = *(const v2f*)g;
#endif
}

__device__ __forceinline__ v8f wmma_f32(v2f a, v2f b, v8f c) {
  // V_WMMA_F32_16X16X4_F32 : D = A(16x4) * B(4x16) + C(16x16)
  return __builtin_amdgcn_wmma_f32_16x16x4_f32(
      false, a, false, b, (short)0, c, false, false);
}

// Generic  out[M,N] = A[M,K] @ W[wrow0+N, K]^T (+bias) (+sigmoid)
// One wave computes a 16x64 tile; 1-deep software pipeline on the K loop so
// next-step global_load_b64s issue before the current 4 WMMAs.
template<int KK>
__global__ void gemm_wmma(const float* __restrict__ A, int lda,
                          const float* __restrict__ W, int ldw, int wrow0,
                          const float* __restrict__ bias, int boff,
                          float* __restrict__ out, int ldo,
                          int nTilesN, int act)
{
  const int wid  = (blockIdx.x * blockDim.x + threadIdx.x) >> 5;
  const int lane = threadIdx.x & 31;
  const int tM = wid / nTilesN;
  const int tN = wid % nTilesN;
  const int hi = lane >> 4;      // half-wave select (K split per ISA layout)
  const int lm = lane & 15;

  const int m = tM * 16 + lm;
  const float* arow = A + (size_t)m * lda + 2 * hi;
  const float* wr[4];
#pragma unroll
  for (int j = 0; j < 4; ++j)
    wr[j] = W + (size_t)(wrow0 + tN * 64 + j * 16 + lm) * ldw + 2 * hi;

  v8f zero = {};
  v8f acc[4] = {zero, zero, zero, zero};

  v2f a_c = *(const v2f*)(arow);
  v2f b_c[4];
#pragma unroll
  for (int j = 0; j < 4; ++j) b_c[j] = *(const v2f*)(wr[j]);

#pragma unroll 2
  for (int k = 0; k < KK - 4; k += 4) {
    v2f a_n = *(const v2f*)(arow + k + 4);
    v2f b_n[4];
#pragma unroll
    for (int j = 0; j < 4; ++j) b_n[j] = *(const v2f*)(wr[j] + k + 4);
#pragma unroll
    for (int j = 0; j < 4; ++j) acc[j] = wmma_f32(a_c, b_c[j], acc[j]);
    a_c = a_n;
#pragma unroll
    for (int j = 0; j < 4; ++j) b_c[j] = b_n[j];
  }
#pragma unroll
  for (int j = 0; j < 4; ++j) acc[j] = wmma_f32(a_c, b_c[j], acc[j]);

#pragma unroll
  for (int j = 0; j < 4; ++j) {
    const int n = tN * 64 + j * 16 + lm;
    const float bv = bias ? bias[boff + n] : 0.0f;
#pragma unroll
    for (int e = 0; e < 8; ++e) {
      const int mm = tM * 16 + e + 8 * hi;     // C/D layout: M = vgpr + 8*hi
      float v = acc[j][e] + bv;
      if (act == 1) v = 1.0f / (1.0f + __expf(-v));
      out[(size_t)mm * ldo + n] = v;
    }
  }
}

// xc = gelu_exact(xp[:, :512]) + conv_b + sum_tap xp_shift[:, 512:] @ conv_w[:,:,tap]^T
__global__ void conv_xc_kernel(const float* __restrict__ xp,   // (512, 1024)
                               const float* __restrict__ Wc,   // (512, 512, 3)
                               const float* __restrict__ cb,   // (512)
                               float* __restrict__ xc)         // (512, 512)
{
  const int wid  = (blockIdx.x * blockDim.x + threadIdx.x) >> 5;
  const int lane = threadIdx.x & 31;
  const int tM = wid >> 3;          // N/64 = 8 tiles
  const int tN = wid & 7;
  const int hi = lane >> 4;
  const int lm = lane & 15;

  const int m   = tM * 16 + lm;
  const int bat = m >> 8;
  const int pos = m & 255;

  const float* wb[4];
#pragma unroll
  for (int j = 0; j < 4; ++j)
    wb[j] = Wc + (size_t)(tN * 64 + j * 16 + lm) * IB * 3 + 2 * hi * 3;

  v8f zero = {};
  v8f acc[4] = {zero, zero, zero, zero};

  for (int tap = 0; tap < 3; ++tap) {
    const int p2 = pos + tap - 1;                 // same-padding shift along L
    const bool valid = (p2 >= 0) && (p2 < LB);
    const float* arow = xp + (size_t)((bat << 8) + p2) * 1024 + 512 + 2 * hi;

    v2f a_c;
    if (valid) a_c = *(const v2f*)(arow);
    else       { a_c.x = 0.0f; a_c.y = 0.0f; }
    v2f b_c[4];
#pragma unroll
    for (int j = 0; j < 4; ++j) {
      b_c[j].x = wb[j][tap];
      b_c[j].y = wb[j][3 + tap];
    }

#pragma unroll 2
    for (int k = 0; k < IB - 4; k += 4) {
      v2f a_n;
      if (valid) a_n = *(const v2f*)(arow + k + 4);
      else       { a_n.x = 0.0f; a_n.y = 0.0f; }
      v2f b_n[4];
#pragma unroll
      for (int j = 0; j < 4; ++j) {
        b_n[j].x = wb[j][(size_t)(k + 4) * 3 + tap];
        b_n[j].y = wb[j][(size_t)(k + 5) * 3 + tap];
      }
#pragma unroll
      for (int j = 0; j < 4; ++j) acc[j] = wmma_f32(a_c, b_c[j], acc[j]);
      a_c = a_n;
#pragma unroll
      for (int j = 0; j < 4; ++j) b_c[j] = b_n[j];
    }
#pragma unroll
    for (int j = 0; j < 4; ++j) acc[j] = wmma_f32(a_c, b_c[j], acc[j]);
  }

#pragma unroll
  for (int j = 0; j < 4; ++j) {
    const int n = tN * 64 + j * 16 + lm;
    const float bv = cb[n];
#pragma unroll
    for (int e = 0; e < 8; ++e) {
      const int mm = tM * 16 + e + 8 * hi;
      const float xq = xp[(size_t)mm * 1024 + n];
      const float ge = 0.5f * xq * (1.0f + erff(xq * 0.70710678118654752f));
      xc[(size_t)mm * IB + n] = acc[j][e] + bv + ge;
    }
  }
}

// xout = xc * dw^2 ; cwgt = g * cwr
__global__ void ew_kernel(const float* __restrict__ xc, const float* __restrict__ dw,
                          const float* __restrict__ g,  const float* __restrict__ cwr,
                          float* __restrict__ xout, float* __restrict__ cwgt)
{
  const int idx = blockIdx.x * blockDim.x + threadIdx.x;
  const float d = dw[idx];
  xout[idx] = xc[idx] * d * d;
  cwgt[idx] = g[idx] * cwr[idx];
}

// Fused selective scan + C contraction + D skip.
// Block = 256 threads = 8 waves = 8 consecutive i of one batch. u/C_weight rows
// for each timestep are staged once per block in LDS (async, double-buffered
// two steps ahead), instead of 8x redundant global reads.
__global__ void scan_kernel(const float* __restrict__ xout,  // (B,L,I) = u
                            const float* __restrict__ cw,    // (B,L,S)
                            const float* __restrict__ Amat,  // (S,I)
                            const float* __restrict__ Dvec,  // (I)
                            float* __restrict__ y)           // (B,L,I)
{
  __shared__ __align__(16) float su[2][IB];
  __shared__ __align__(16) float sc[2][IB];

  const int tid  = threadIdx.x;
  const int lane = tid & 31;
  const int wv   = tid >> 5;                  // 0..7
  const int bat  = blockIdx.x >> 6;           // 128 blocks: 64 per batch
  const int i    = ((blockIdx.x & 63) << 3) + wv;

  // per-lane states: s = 2*lane + 64*j (+1), j = 0..7  -> b64 LDS reads
  float a[16], h[16];
#pragma unroll
  for (int j = 0; j < 8; ++j) {
    const int s0 = 2 * lane + 64 * j;
    a[2 * j]     = Amat[(size_t)s0 * IB + i];
    a[2 * j + 1] = Amat[(size_t)(s0 + 1) * IB + i];
    h[2 * j] = 0.0f;
    h[2 * j + 1] = 0.0f;
  }
  const float Di = Dvec[i];
  const float* ub = xout + (size_t)bat * LB * IB;
  const float* cb = cw   + (size_t)bat * LB * IB;
  float*       yb = y    + (size_t)bat * LB * IB;
  const int e2 = tid * 2;                     // 2 floats per thread per array

#if USE_ASYNC_LDS
  // prologue: prefetch t=0 and t=1 (4 async ops per wave outstanding)
  async_copy_b64(ub + e2,      &su[0][e2]);
  async_copy_b64(cb + e2,      &sc[0][e2]);
  async_copy_b64(ub + IB + e2, &su[1][e2]);
  async_copy_b64(cb + IB + e2, &sc[1][e2]);
#endif

  for (int t = 0; t < LB; ++t) {
    const int cur = t & 1;
#if USE_ASYNC_LDS
    __builtin_amdgcn_s_wait_asynccnt(2);      // this step's 2 ops complete (in-order)
    __syncthreads();                          // all waves' slices landed
#else
    *(v2f*)&su[cur][e2] = *(const v2f*)(ub + (size_t)t * IB + e2);
    *(v2f*)&sc[cur][e2] = *(const v2f*)(cb + (size_t)t * IB + e2);
    __syncthreads();
#endif

    float partial = 0.0f;
#pragma unroll
    for (int j = 0; j < 8; ++j) {
      const int s0 = 2 * lane + 64 * j;       // conflict-free b64 pattern
      const v2f uu = *(const v2f*)&su[cur][s0];
      const v2f cc = *(const v2f*)&sc[cur][s0];
      h[2 * j]     = fmaf(a[2 * j],     h[2 * j],     uu.x);
      h[2 * j + 1] = fmaf(a[2 * j + 1], h[2 * j + 1], uu.y);
      partial = fmaf(h[2 * j],     cc.x, partial);
      partial = fmaf(h[2 * j + 1], cc.y, partial);
    }
#pragma unroll
    for (int off = 16; off > 0; off >>= 1)
      partial += __shfl_xor(partial, off, 32);
    if (lane == 0)
      yb[(size_t)t * IB + i] = partial + su[cur][i] * Di;

    __syncthreads();                          // everyone done reading buf `cur`
#if USE_ASYNC_LDS
    if (t + 2 < LB) {                         // refill buf `cur` with step t+2
      async_copy_b64(ub + (size_t)(t + 2) * IB + e2, &su[cur][e2]);
      async_copy_b64(cb + (size_t)(t + 2) * IB + e2, &sc[cur][e2]);
    }
#endif
  }
}

// Row LayerNorm over H=256, * z, + residual x. One wave per row.
__global__ void ln_kernel(const float* __restrict__ op, const float* __restrict__ x,
                          const float* __restrict__ lg, const float* __restrict__ lb,
                          const float* __restrict__ z,  float* __restrict__ out)
{
  const int row  = (blockIdx.x * blockDim.x + threadIdx.x) >> 5;  // 0..511
  const int lane = threadIdx.x & 31;
  const float* r = op + (size_t)row * HB;
  float v[8];
  float s = 0.0f, ss = 0.0f;
#pragma unroll
  for (int e = 0; e < 8; ++e) {
    v[e] = r[lane + 32 * e];
    s += v[e];
    ss += v[e] * v[e];
  }
#pragma unroll
  for (int off = 16; off > 0; off >>= 1) {
    s  += __shfl_xor(s,  off, 32);
    ss += __shfl_xor(ss, off, 32);
  }
  const float mu  = s * (1.0f / HB);
  const float var = ss * (1.0f / HB) - mu * mu;
  const float inv = rsqrtf(var + 1e-5f);
  const float zz  = z[0];
#pragma unroll
  for (int e = 0; e < 8; ++e) {
    const int c = lane + 32 * e;
    const float o = (v[e] - mu) * inv * lg[c] + lb[c];
    out[(size_t)row * HB + c] = o * zz + x[(size_t)row * HB + c];
  }
}

extern "C" void kernel_launch(void* const* d_in, const int* in_sizes, int n_in,
                              void* d_out, int out_size, void* d_ws, size_t ws_size,
                              hipStream_t stream)
{
  const float* x      = (const float*)d_in[0];
  const float* W_in   = (const float*)d_in[1];
  const float* convw  = (const float*)d_in[2];
  const float* convb  = (const float*)d_in[3];
  const float* W_ssm  = (const float*)d_in[4];
  const float* b_ssm  = (const float*)d_in[5];
  const float* W_gate = (const float*)d_in[6];
  const float* b_gate = (const float*)d_in[7];
  const float* Amat   = (const float*)d_in[8];
  const float* Dvec   = (const float*)d_in[9];
  const float* W_out  = (const float*)d_in[10];
  const float* z      = (const float*)d_in[11];
  const float* ln_g   = (const float*)d_in[12];
  const float* ln_b   = (const float*)d_in[13];
  float* out = (float*)d_out;

  float* ws   = (float*)d_ws;
  float* xp   = ws;                  // 512*1024
  float* xc   = xp   + 512 * 1024;   // 512*512
  float* cwr  = xc   + 512 * 512;
  float* dw   = cwr  + 512 * 512;
  float* g    = dw   + 512 * 512;
  float* xout = g    + 512 * 512;
  float* cwgt = xout + 512 * 512;
  float* y    = cwgt + 512 * 512;
  float* opre = y    + 512 * 512;    // 512*256   (total ~10 MB)

  // 1) xp = x @ W_in^T : 512 waves (32 Mtiles x 16 Ntiles)
  gemm_wmma<256><<<128, 128, 0, stream>>>(x, HB, W_in, HB, 0, nullptr, 0,
                                          xp, 1024, 16, 0);
  // 2) xc = gelu(x_q) + conv(x_k) + conv_b : 256 waves
  conv_xc_kernel<<<64, 128, 0, stream>>>(xp, convw, convb, xc);
  // 3) only the used slices of ssm/gate : 256 waves each
  gemm_wmma<512><<<64, 128, 0, stream>>>(xc, IB, W_ssm, IB, 512,  b_ssm, 512,
                                         cwr, IB, 8, 0);
  gemm_wmma<512><<<64, 128, 0, stream>>>(xc, IB, W_ssm, IB, 1024, b_ssm, 1024,
                                         dw, IB, 8, 0);
  gemm_wmma<512><<<64, 128, 0, stream>>>(xc, IB, W_gate, IB, 512, b_gate, 512,
                                         g, IB, 8, 1);
  // 4) elementwise prep (262144 elements)
  ew_kernel<<<1024, 256, 0, stream>>>(xc, dw, g, cwr, xout, cwgt);
  // 5) fused scan + contract : 128 blocks x 8 waves (one (b,i) per wave)
  scan_kernel<<<128, 256, 0, stream>>>(xout, cwgt, Amat, Dvec, y);
  // 6) opre = y @ W_out^T : 128 waves (32 Mtiles x 4 Ntiles)
  gemm_wmma<512><<<32, 128, 0, stream>>>(y, IB, W_out, IB, 0, nullptr, 0,
                                         opre, HB, 4, 0);
  // 7) layernorm + z + residual : 512 waves
  ln_kernel<<<64, 256, 0, stream>>>(opre, x, ln_g, ln_b, z, out);
}